// SGCN_6871947673680
// MI455X (gfx1250) — compile-verified
//
#include <hip/hip_runtime.h>
#include <cmath>

// ---------------- problem constants (from reference) ----------------
#define NNODES 100000
#define DIM    64
#define LL0    64
#define LL1    32
#define EPOS   400000
#define ENEG   400000

typedef __attribute__((ext_vector_type(2))) float v2f;
typedef __attribute__((ext_vector_type(8))) float v8f;

__device__ __forceinline__ void atomic_add_f32(float* p, float v) {
  unsafeAtomicAdd(p, v);   // hardware global_atomic_add_f32
}

// ---------------- utility kernels ----------------
__global__ void zero_kernel(float* __restrict__ p, long n) {
  long i = (long)blockIdx.x * blockDim.x + threadIdx.x;
  long stride = (long)gridDim.x * blockDim.x;
  for (; i < n; i += stride) p[i] = 0.0f;
}

__global__ void copy_kernel(const float* __restrict__ src, float* __restrict__ dst, long n) {
  long i = (long)blockIdx.x * blockDim.x + threadIdx.x;
  long stride = (long)gridDim.x * blockDim.x;
  for (; i < n; i += stride) dst[i] = src[i];
}

// counts[rows[e]] += 1
__global__ void count_kernel(const int* __restrict__ rows, float* __restrict__ counts, int E) {
  int e = blockIdx.x * blockDim.x + threadIdx.x;
  if (e < E) atomic_add_f32(&counts[rows[e]], 1.0f);
}

// agg[row[e], f] += src[col[e], f]   (one thread per (edge, feature))
__global__ void scatter_add_kernel(const float* __restrict__ src,
                                   const int* __restrict__ rows,
                                   const int* __restrict__ cols,
                                   float* __restrict__ agg, int E) {
  long idx = (long)blockIdx.x * blockDim.x + threadIdx.x;
  if (idx >= (long)E * DIM) return;
  int e = (int)(idx >> 6);
  int f = (int)(idx & 63);
  long r = rows[e];
  long c = cols[e];
  atomic_add_f32(&agg[r * DIM + f], src[c * DIM + f]);
}

// agg[n, f] /= max(counts[n], 1)
__global__ void normalize_kernel(float* __restrict__ agg, const float* __restrict__ counts) {
  long idx = (long)blockIdx.x * blockDim.x + threadIdx.x;
  if (idx >= (long)NNODES * DIM) return;
  float c = counts[idx >> 6];
  agg[idx] = agg[idx] / fmaxf(c, 1.0f);
}

// ---------------- fp32 WMMA GEMM: Out[:, colOff:colOff+Nout] = tanh(concat(segs) @ W + b)
// Each segment is row-major [M x 64]; K = nseg*64; W is row-major [K x Nout] in memory
// but staged TRANSPOSED into LDS (sWT[n*K + k]) so each lane's K-pair is one ds_load_b64.
// One wave32 computes a 16x16 tile via V_WMMA_F32_16X16X4_F32, dual accumulators for ILP.
__global__ __launch_bounds__(256) void wmma_gemm_tanh(
    const float* __restrict__ segA0, const float* __restrict__ segA1,
    const float* __restrict__ segA2, int nseg,
    const float* __restrict__ W, const float* __restrict__ bias,
    float* __restrict__ Out, int ldOut, int colOff, int Nout, int Mtiles)
{
  __shared__ float sWT[8192];                   // up to 128x64 / 192x32 fp32, transposed
  const int K = nseg * 64;
  for (int i = threadIdx.x; i < K * Nout; i += blockDim.x) {
    const int k = i / Nout;
    const int n = i - k * Nout;
    sWT[n * K + k] = W[i];
  }
  __syncthreads();

  const int wave   = threadIdx.x >> 5;
  const int lane   = threadIdx.x & 31;
  const int ntiles = Nout >> 4;
  const int tile   = blockIdx.x * (blockDim.x >> 5) + wave;
  if (tile >= Mtiles * ntiles) return;          // wave-uniform: EXEC stays all-1s
  const int mt = tile / ntiles;
  const int nt = tile - mt * ntiles;

  const int half = lane >> 4;                   // 0: K pair {0,1}, 1: K pair {2,3}
  const int l16  = lane & 15;
  const long m   = (long)mt * 16 + l16;         // A row held by this lane
  const int  n   = nt * 16 + l16;               // B/C column held by this lane
  const int  kh  = half * 2;

  const float* bcol = sWT + (long)n * K + kh;   // lane's column, base K offset

  v8f c0 = {}, c1 = {};                         // dual accumulators -> 2-deep WMMA pipe
  for (int sg = 0; sg < nseg; ++sg) {
    const float* seg  = (sg == 0) ? segA0 : ((sg == 1) ? segA1 : segA2);
    const float* arow = seg + m * 64 + kh;
    const float* bseg = bcol + sg * 64;
#pragma unroll
    for (int s = 0; s < 16; ++s) {              // 64 K-values per segment, 4 per step
      const int k = s << 2;                     // even -> 8B-aligned float2 loads
      const v2f a = *(const v2f*)(arow + k);    // global_load_b64
      const v2f b = *(const v2f*)(bseg + k);    // ds_load_b64
      if (s & 1)
        c1 = __builtin_amdgcn_wmma_f32_16x16x4_f32(false, a, false, b,
                                                   (short)0, c1, false, false);
      else
        c0 = __builtin_amdgcn_wmma_f32_16x16x4_f32(false, a, false, b,
                                                   (short)0, c0, false, false);
    }
  }

  const float bv = bias[n];
#pragma unroll
  for (int v = 0; v < 8; ++v) {                 // C layout: VGPR v -> row v + 8*half
    const long row = (long)mt * 16 + v + 8 * half;
    Out[row * ldOut + colOff + n] = tanhf(c0[v] + c1[v] + bv);
  }
}

// ---------------- triplet loss: sum over edges of relu(|zi-zj|^2 - |zj-zk|^2)
__global__ void triplet_kernel(const float* __restrict__ z,
                               const int* __restrict__ edges,   // [2, E] flat
                               const int* __restrict__ samples,
                               int E, float* __restrict__ accum)
{
  const int e = blockIdx.x * blockDim.x + threadIdx.x;
  float val = 0.0f;
  if (e < E) {
    const long i = edges[e];
    const long j = edges[E + e];
    const long k = samples[e];
    const float4* zi = (const float4*)(z + i * 64);
    const float4* zj = (const float4*)(z + j * 64);
    const float4* zk = (const float4*)(z + k * 64);
    float nij = 0.0f, nik = 0.0f;
#pragma unroll
    for (int t = 0; t < 16; ++t) {
      const float4 a = zi[t], b = zj[t], cc = zk[t];
      float dx = a.x - b.x, dy = a.y - b.y, dz = a.z - b.z, dw = a.w - b.w;
      nij += dx * dx + dy * dy + dz * dz + dw * dw;
      float ex = b.x - cc.x, ey = b.y - cc.y, ez = b.z - cc.z, ew = b.w - cc.w;
      nik += ex * ex + ey * ey + ez * ez + ew * ew;
    }
    val = fmaxf(nij - nik, 0.0f);
  }
  for (int off = 16; off > 0; off >>= 1) val += __shfl_down(val, off, 32);
  if ((threadIdx.x & 31) == 0) atomic_add_f32(accum, val);
}

// ---------------- regression: sum over rows of -(logp[target])
__global__ void regress_kernel(const float* __restrict__ z,
                               const int* __restrict__ pe,
                               const int* __restrict__ ne,
                               const int* __restrict__ target,
                               const float* __restrict__ Wreg,   // [128, 2]
                               float* __restrict__ accum)
{
  __shared__ float sWr[256];
  for (int i = threadIdx.x; i < 256; i += blockDim.x) sWr[i] = Wreg[i];
  __syncthreads();

  const long r = (long)blockIdx.x * blockDim.x + threadIdx.x;
  float val = 0.0f;
  if (r < (long)(EPOS + ENEG)) {
    long i, j;
    if (r < EPOS) { i = pe[r]; j = pe[EPOS + r]; }
    else          { long rr = r - EPOS; i = ne[rr]; j = ne[ENEG + rr]; }
    const float* zi = z + i * 64;
    const float* zj = z + j * 64;
    float l0 = 0.0f, l1 = 0.0f;
#pragma unroll 8
    for (int f = 0; f < 64; ++f) {
      const float a = zi[f], b = zj[f];
      l0 += a * sWr[f * 2 + 0] + b * sWr[(64 + f) * 2 + 0];
      l1 += a * sWr[f * 2 + 1] + b * sWr[(64 + f) * 2 + 1];
    }
    const float mx  = fmaxf(l0, l1);
    const float lse = mx + logf(expf(l0 - mx) + expf(l1 - mx));
    const float lt  = (target[r] == 0) ? l0 : l1;
    val = lse - lt;                       // = -log_softmax[target]
  }
  for (int off = 16; off > 0; off >>= 1) val += __shfl_down(val, off, 32);
  if ((threadIdx.x & 31) == 0) atomic_add_f32(accum, val);
}

__global__ void finalize_kernel(const float* __restrict__ accum, float* __restrict__ out) {
  out[0] = accum[0] / (float)(EPOS + ENEG)
         + accum[1] / (float)EPOS
         + accum[2] / (float)ENEG;
}

// ---------------- host-side orchestration ----------------
extern "C" void kernel_launch(void* const* d_in, const int* in_sizes, int n_in,
                              void* d_out, int out_size, void* d_ws, size_t ws_size,
                              hipStream_t stream) {
  const float* X      = (const float*)d_in[0];
  const float* W_pos0 = (const float*)d_in[1];
  const float* b_pos0 = (const float*)d_in[2];
  const float* W_neg0 = (const float*)d_in[3];
  const float* b_neg0 = (const float*)d_in[4];
  const float* W_pos1 = (const float*)d_in[5];
  const float* b_pos1 = (const float*)d_in[6];
  const float* W_neg1 = (const float*)d_in[7];
  const float* b_neg1 = (const float*)d_in[8];
  const float* W_reg  = (const float*)d_in[9];
  const int*   pe     = (const int*)d_in[10];   // [2, EPOS]
  const int*   ne     = (const int*)d_in[11];   // [2, ENEG]
  const int*   target = (const int*)d_in[12];
  const int*   pos_s  = (const int*)d_in[13];
  const int*   neg_s  = (const int*)d_in[14];
  float* out = (float*)d_out;                   // [0] = loss, [1..] = z [N, 64]

  const long ZN = (long)NNODES * 64;
  float* w     = (float*)d_ws;
  float* zws   = w;                 // aligned copy of z
  float* agg_a = zws + ZN;
  float* agg_b = agg_a + ZN;
  float* hpos  = agg_b + ZN;
  float* hneg  = hpos + ZN;
  float* cpos  = hneg + ZN;         // [NNODES]
  float* cneg  = cpos + NNODES;     // [NNODES]
  float* accum = cneg + NNODES;     // [0]=regress sum, [1]=pos sum, [2]=neg sum

  const int T = 256;
  const int gbE   = (EPOS + T - 1) / T;                   // per-edge
  const int gbEF  = (int)(((long)EPOS * 64 + T - 1) / T); // per-(edge,feature)
  const int gbNF  = (int)((ZN + T - 1) / T);              // per-(node,feature)
  const int gbG0  = (6250 * 4 + 7) / 8;                   // layer-0 GEMM tiles / 8 waves
  const int gbG1  = (6250 * 2 + 7) / 8;                   // layer-1 GEMM tiles / 8 waves
  const int gbReg = (int)(((long)(EPOS + ENEG) + T - 1) / T);

  // 0) zero accumulators, counts, agg buffers
  zero_kernel<<<4096, T, 0, stream>>>(agg_a, 2 * ZN);
  zero_kernel<<<1024, T, 0, stream>>>(cpos, 2L * NNODES + 8);

  // 1) per-destination edge counts (reused by all layers)
  count_kernel<<<gbE, T, 0, stream>>>(pe, cpos, EPOS);
  count_kernel<<<gbE, T, 0, stream>>>(ne, cneg, ENEG);

  // 2) base layer scatter-means of X
  scatter_add_kernel<<<gbEF, T, 0, stream>>>(X, pe, pe + EPOS, agg_a, EPOS);
  scatter_add_kernel<<<gbEF, T, 0, stream>>>(X, ne, ne + ENEG, agg_b, ENEG);
  normalize_kernel<<<gbNF, T, 0, stream>>>(agg_a, cpos);
  normalize_kernel<<<gbNF, T, 0, stream>>>(agg_b, cneg);

  // 3) base layer GEMMs: h = tanh([agg | X] @ W0 + b0)
  wmma_gemm_tanh<<<gbG0, T, 0, stream>>>(agg_a, X, nullptr, 2, W_pos0, b_pos0,
                                         hpos, LL0, 0, LL0, 6250);
  wmma_gemm_tanh<<<gbG0, T, 0, stream>>>(agg_b, X, nullptr, 2, W_neg0, b_neg0,
                                         hneg, LL0, 0, LL0, 6250);

  // 4) deep layer (pos): agg1 = mean(hpos over pos), agg2 = mean(hneg over neg)
  zero_kernel<<<4096, T, 0, stream>>>(agg_a, 2 * ZN);
  scatter_add_kernel<<<gbEF, T, 0, stream>>>(hpos, pe, pe + EPOS, agg_a, EPOS);
  scatter_add_kernel<<<gbEF, T, 0, stream>>>(hneg, ne, ne + ENEG, agg_b, ENEG);
  normalize_kernel<<<gbNF, T, 0, stream>>>(agg_a, cpos);
  normalize_kernel<<<gbNF, T, 0, stream>>>(agg_b, cneg);
  wmma_gemm_tanh<<<gbG1, T, 0, stream>>>(agg_a, agg_b, hpos, 3, W_pos1, b_pos1,
                                         zws, 64, 0, LL1, 6250);   // z[:, 0:32]

  // 5) deep layer (neg): agg1 = mean(hneg over pos), agg2 = mean(hpos over neg)
  zero_kernel<<<4096, T, 0, stream>>>(agg_a, 2 * ZN);
  scatter_add_kernel<<<gbEF, T, 0, stream>>>(hneg, pe, pe + EPOS, agg_a, EPOS);
  scatter_add_kernel<<<gbEF, T, 0, stream>>>(hpos, ne, ne + ENEG, agg_b, ENEG);
  normalize_kernel<<<gbNF, T, 0, stream>>>(agg_a, cpos);
  normalize_kernel<<<gbNF, T, 0, stream>>>(agg_b, cneg);
  wmma_gemm_tanh<<<gbG1, T, 0, stream>>>(agg_a, agg_b, hneg, 3, W_neg1, b_neg1,
                                         zws, 64, 32, LL1, 6250);  // z[:, 32:64]

  // 6) losses
  triplet_kernel<<<gbE, T, 0, stream>>>(zws, pe, pos_s, EPOS, accum + 1);
  triplet_kernel<<<gbE, T, 0, stream>>>(zws, ne, neg_s, ENEG, accum + 2);
  regress_kernel<<<gbReg, T, 0, stream>>>(zws, pe, ne, target, W_reg, accum + 0);

  // 7) outputs
  finalize_kernel<<<1, 1, 0, stream>>>(accum, out);
  copy_kernel<<<4096, T, 0, stream>>>(zws, out + 1, ZN);
}